// ODEfunc_23063974379975
// MI455X (gfx1250) — compile-verified
//
#include <hip/hip_runtime.h>
#include <math.h>

// ---------------------------------------------------------------------------
// ODE func + analytic divergence for MI455X (gfx1250, wave32, WMMA fp32)
//   h   = tanh(y @ W1 + t*wt + b1)           [N,H]
//   dy  = h @ W2 + b2                        [N,D]
//   div = sum_k (1 - h_k^2) * s_k,  s_k = sum_i W1[i,k]*W2[k,i]
//   out = [dy | -div]                        [N,D+1]
// ---------------------------------------------------------------------------

typedef __attribute__((ext_vector_type(2))) float v2f;
typedef __attribute__((ext_vector_type(8))) float v8f;

#define NROWS 8192
#define DDIM  64
#define HDIM  256

#define W1T_STRIDE 66    // 64 k-values + 2 pad  (bank decorrelation)
#define W2T_STRIDE 258   // 256 k-values + 2 pad
#define HBUF_STRIDE 258  // 256 cols + 2 pad

#define W1T_ELEMS (HDIM * W1T_STRIDE)   // 16896 floats
#define W2T_ELEMS (DDIM * W2T_STRIDE)   // 16512 floats
#define HBUF_ELEMS (16 * HBUF_STRIDE)   // 4128 floats per wave

#define WAVES_PER_BLOCK 4
#define BLOCK_THREADS (WAVES_PER_BLOCK * 32)
#define ROWS_PER_WAVE 16
#define ROWS_PER_BLOCK (WAVES_PER_BLOCK * ROWS_PER_WAVE)  // 64

#define SMEM_FLOATS (W1T_ELEMS + W2T_ELEMS + WAVES_PER_BLOCK * HBUF_ELEMS)
#define SMEM_BYTES  (SMEM_FLOATS * 4)   // 199680 B < 320KB/WG on CDNA5

// Branchless tanh on HW transcendentals: v_exp_f32 + v_rcp_f32 + fma + bfi.
// Never touches EXEC (keeps the wave WMMA-legal, no saveexec/branch bloat).
// exp2 -> +inf for large |x| gives exact +/-1 saturation.
__device__ __forceinline__ float fast_tanhf(float x) {
  const float ax = __builtin_fabsf(x);
  const float e  = __builtin_amdgcn_exp2f(ax * 2.885390081777927f); // 2*log2(e)
  const float r  = 1.0f - 2.0f * __builtin_amdgcn_rcpf(e + 1.0f);
  return __builtin_copysignf(r, x);
}

// --- prep: s_k and bias_k (one 256-thread block) ---------------------------
__global__ void ode_prep_kernel(const float* __restrict__ t,
                                const float* __restrict__ W1,
                                const float* __restrict__ b1,
                                const float* __restrict__ wt,
                                const float* __restrict__ W2,
                                float* __restrict__ ws) {
  const int k = threadIdx.x;  // 0..255
  float s = 0.f;
#pragma unroll
  for (int i = 0; i < DDIM; ++i) s += W1[i * HDIM + k] * W2[k * DDIM + i];
  ws[k]        = s;                       // s[256]
  ws[HDIM + k] = t[0] * wt[k] + b1[k];    // bias[256]
}

// --- main: two fp32 WMMA GEMMs + tanh + divergence -------------------------
__global__ void __launch_bounds__(BLOCK_THREADS)
ode_main_kernel(const float* __restrict__ y,
                const float* __restrict__ W1,
                const float* __restrict__ W2,
                const float* __restrict__ b2,
                const float* __restrict__ sbias,  // ws: s[256] | bias[256]
                float* __restrict__ out) {
  extern __shared__ float smem[];
  float* w1t = smem;                       // W1^T [256][66]
  float* w2t = smem + W1T_ELEMS;           // W2^T [64][258]
  const int tid  = threadIdx.x;
  const int wave = tid >> 5;
  const int lane = tid & 31;
  float* hbuf = smem + W1T_ELEMS + W2T_ELEMS + wave * HBUF_ELEMS;  // [16][258]

  // Stage W1 transposed: w1t[n][i] = W1[i][n]   (coalesced global reads)
  for (int e = tid; e < DDIM * HDIM; e += BLOCK_THREADS) {
    const int i = e >> 8;      // row of W1 (k-dim of GEMM1)
    const int n = e & 255;     // col of W1
    w1t[n * W1T_STRIDE + i] = W1[e];
  }
  // Stage W2 transposed: w2t[n][k] = W2[k][n]
  for (int e = tid; e < HDIM * DDIM; e += BLOCK_THREADS) {
    const int k = e >> 6;
    const int n = e & 63;
    w2t[n * W2T_STRIDE + k] = W2[e];
  }
  __syncthreads();

  const int m16   = lane & 15;
  const int hi    = lane >> 4;   // half-wave select
  const int khalf = hi * 2;      // K sub-offset per 16x16x4 layout
  const int rowBase = blockIdx.x * ROWS_PER_BLOCK + wave * ROWS_PER_WAVE;

  // A fragments for GEMM1: lane holds y[rowBase+m16][4*kk+khalf .. +1]
  const float* yrow = y + (size_t)(rowBase + m16) * DDIM;
  v2f afrag[16];
#pragma unroll
  for (int kk = 0; kk < 16; ++kk)
    afrag[kk] = *(const v2f*)(yrow + kk * 4 + khalf);

  const float* svec = sbias;
  const float* bias = sbias + HDIM;

  float divacc[8];
#pragma unroll
  for (int r = 0; r < 8; ++r) divacc[r] = 0.f;

  // ---- GEMM1 (16x256 = 16 tiles x 16 K-steps) + tanh/div epilogue ----
  for (int t = 0; t < 16; ++t) {
    v8f acc = {};
    const int col = t * 16 + m16;
    const float* bcol = w1t + col * W1T_STRIDE;
#pragma unroll
    for (int kk = 0; kk < 16; ++kk) {
      const v2f bfrag = *(const v2f*)(bcol + kk * 4 + khalf);
      acc = __builtin_amdgcn_wmma_f32_16x16x4_f32(
          false, afrag[kk], false, bfrag, (short)0, acc, false, false);
    }
    const float bv = bias[col];
    const float sk = svec[col];
#pragma unroll
    for (int r = 0; r < 8; ++r) {
      const float hval = fast_tanhf(acc[r] + bv);
      divacc[r] = __builtin_fmaf(-hval * hval + 1.f, sk, divacc[r]);
      hbuf[(r + hi * 8) * HBUF_STRIDE + col] = hval;  // C-layout -> row-major
    }
  }

  // ---- GEMM2: dy = h @ W2  (16x64 = 4 tiles, K=256 = 64 steps) ----
  v8f zero = {};
  v8f acc2[4] = {zero, zero, zero, zero};
  const float* harow = hbuf + m16 * HBUF_STRIDE;
  for (int kk = 0; kk < 64; ++kk) {
    const v2f a2 = *(const v2f*)(harow + kk * 4 + khalf);
#pragma unroll
    for (int t2 = 0; t2 < 4; ++t2) {
      const v2f bf = *(const v2f*)(w2t + (t2 * 16 + m16) * W2T_STRIDE + kk * 4 + khalf);
      acc2[t2] = __builtin_amdgcn_wmma_f32_16x16x4_f32(
          false, a2, false, bf, (short)0, acc2[t2], false, false);
    }
  }

  // dy epilogue: add b2, write out[row][col] with row stride D+1
#pragma unroll
  for (int t2 = 0; t2 < 4; ++t2) {
    const int col = t2 * 16 + m16;
    const float bb = b2[col];
#pragma unroll
    for (int r = 0; r < 8; ++r) {
      const int row = rowBase + r + hi * 8;
      out[(size_t)row * (DDIM + 1) + col] = acc2[t2][r] + bb;
    }
  }

  // divergence: reduce 16 lanes per half-wave (xor<16 stays within halves)
#pragma unroll
  for (int r = 0; r < 8; ++r) {
    float v = divacc[r];
    v += __shfl_xor(v, 8, 32);
    v += __shfl_xor(v, 4, 32);
    v += __shfl_xor(v, 2, 32);
    v += __shfl_xor(v, 1, 32);
    if (m16 == 0) {
      const int row = rowBase + r + hi * 8;
      out[(size_t)row * (DDIM + 1) + DDIM] = -v;
    }
  }
}

extern "C" void kernel_launch(void* const* d_in, const int* in_sizes, int n_in,
                              void* d_out, int out_size, void* d_ws, size_t ws_size,
                              hipStream_t stream) {
  (void)in_sizes; (void)n_in; (void)out_size; (void)ws_size;
  const float* t  = (const float*)d_in[0];
  const float* y  = (const float*)d_in[1];
  const float* W1 = (const float*)d_in[2];
  const float* b1 = (const float*)d_in[3];
  const float* wt = (const float*)d_in[4];
  const float* W2 = (const float*)d_in[5];
  const float* b2 = (const float*)d_in[6];
  float* out = (float*)d_out;
  float* ws  = (float*)d_ws;   // s[256] | bias[256]

  hipFuncSetAttribute(reinterpret_cast<const void*>(ode_main_kernel),
                      hipFuncAttributeMaxDynamicSharedMemorySize, SMEM_BYTES);

  ode_prep_kernel<<<1, HDIM, 0, stream>>>(t, W1, b1, wt, W2, ws);
  ode_main_kernel<<<NROWS / ROWS_PER_BLOCK, BLOCK_THREADS, SMEM_BYTES, stream>>>(
      y, W1, W2, b2, ws, out);
}